// GraphMultiHeadAttLayer_48550310314071
// MI455X (gfx1250) — compile-verified
//
#include <hip/hip_runtime.h>
#include <hip/hip_bf16.h>

typedef __attribute__((ext_vector_type(2))) float v2f;
typedef __attribute__((ext_vector_type(8))) float v8f;

#define GAT_ALPHA 0.2f
#define GAT_D 64

// ---------------------------------------------------------------------------
// Kernel 1: per-node scores via fp32 WMMA.
// scores[n][c], c = {h0_src, h0_dst, h1_src, h1_dst}
// A = x (16-node tile x 64), B[k][c] = W[c/2][k] * a[c/2][(c&1)*64 + k]
// B held in LDS column-major, zero-padded to 16 columns so every lane issues
// one unconditional ds_load_b64 per K-step (no EXEC-masked loads).
// ---------------------------------------------------------------------------
__global__ __launch_bounds__(256) void gat_scores_wmma(
    const float* __restrict__ x, const float* __restrict__ W,
    const float* __restrict__ a, float* __restrict__ s, int nTiles)
{
    __shared__ float Bcol[16][GAT_D];   // [column n][k], cols 4..15 = 0
    const int tid = threadIdx.x;
#pragma unroll
    for (int i = 0; i < 4; ++i) {       // 256 threads fill 1024 entries
        const int idx = tid + 256 * i;
        const int n = idx >> 6, k = idx & 63, hc = n >> 1;
        Bcol[n][k] = (n < 4)
            ? W[hc * GAT_D + k] * a[hc * 2 * GAT_D + (n & 1) * GAT_D + k]
            : 0.0f;
    }
    __syncthreads();

    const int tile = blockIdx.x * 8 + (tid >> 5);   // wave per 16-node tile
    if (tile >= nTiles) return;                     // uniform per wave
    const int lane = tid & 31;
    const int m    = lane & 15;                     // A-row / B,D-column
    const int half = lane >> 4;                     // K sub-slice selector

    const float* arow = x + ((size_t)tile * 16 + m) * GAT_D + 2 * half;
    const float* brow = &Bcol[m][2 * half];
    v8f acc = {};
#pragma unroll
    for (int kb = 0; kb < GAT_D; kb += 4) {
        const v2f av = *(const v2f*)(arow + kb);    // global_load_b64
        const v2f bv = *(const v2f*)(brow + kb);    // ds_load_b64
        acc = __builtin_amdgcn_wmma_f32_16x16x4_f32(
            /*neg_a=*/false, av, /*neg_b=*/false, bv,
            /*c_mod=*/(short)0, acc, /*reuse_a=*/false, /*reuse_b=*/false);
    }
    // D layout: lane holds column m, rows 8*half + v  (v = VGPR index 0..7)
    if (m < 4) {
        size_t base = ((size_t)tile * 16 + 8 * half) * 4 + m;
#pragma unroll
        for (int v = 0; v < 8; ++v) s[base + (size_t)v * 4] = acc[v];
    }
}

// ---------------------------------------------------------------------------
// Kernel 2: zero accumulators (h_prime[2][N][64] + rowsum[2][N])
// ---------------------------------------------------------------------------
__global__ __launch_bounds__(256) void gat_zero(float4* __restrict__ p, size_t n4)
{
    size_t i = (size_t)blockIdx.x * blockDim.x + threadIdx.x;
    size_t stride = (size_t)gridDim.x * blockDim.x;
    float4 z = {0.f, 0.f, 0.f, 0.f};
    for (; i < n4; i += stride) p[i] = z;
}

// ---------------------------------------------------------------------------
// Kernel 3: edge scatter. One wave32 per edge; each lane owns 2 features.
// ---------------------------------------------------------------------------
__global__ __launch_bounds__(256) void gat_edges(
    const float* __restrict__ x, const int* __restrict__ row,
    const int* __restrict__ col, const float* __restrict__ s,
    float* __restrict__ hp, float* __restrict__ rowsum, int E, int N)
{
    const int wave = (blockIdx.x * (blockDim.x >> 5)) + (threadIdx.x >> 5);
    if (wave >= E) return;
    const int lane = threadIdx.x & 31;
    const int r = row[wave];
    const int c = col[wave];

    // per-edge attention weights, both heads (redundant across lanes; cached)
    float sc0 = s[(size_t)r * 4 + 0] + s[(size_t)c * 4 + 1];
    float sc1 = s[(size_t)r * 4 + 2] + s[(size_t)c * 4 + 3];
    sc0 = (sc0 > 0.f) ? sc0 : GAT_ALPHA * sc0;
    sc1 = (sc1 > 0.f) ? sc1 : GAT_ALPHA * sc1;
    const float w0 = expf(sc0);
    const float w1 = expf(sc1);

    if (lane == 0) {
        atomicAdd(&rowsum[r], w0);
        atomicAdd(&rowsum[(size_t)N + r], w1);
    }
    const float2 xv = *(const float2*)(x + (size_t)c * GAT_D + lane * 2);
    float* h0 = hp + (size_t)r * GAT_D + lane * 2;
    float* h1 = hp + (size_t)N * GAT_D + (size_t)r * GAT_D + lane * 2;
    atomicAdd(h0,     w0 * xv.x);
    atomicAdd(h0 + 1, w0 * xv.y);
    atomicAdd(h1,     w1 * xv.x);
    atomicAdd(h1 + 1, w1 * xv.y);
}

// ---------------------------------------------------------------------------
// Kernel 4: out = 0.5 * (hp0/rowsum0 + hp1/rowsum1), float4 per thread
// ---------------------------------------------------------------------------
__global__ __launch_bounds__(256) void gat_finalize(
    const float* __restrict__ hp, const float* __restrict__ rowsum,
    float* __restrict__ out, int N)
{
    const int idx = blockIdx.x * blockDim.x + threadIdx.x; // N*16 threads
    if (idx >= N * 16) return;
    const int node = idx >> 4;
    const int q    = (idx & 15) * 4;
    const float inv0 = 1.0f / rowsum[node];
    const float inv1 = 1.0f / rowsum[(size_t)N + node];
    const float4 h0 = *(const float4*)(hp + (size_t)node * GAT_D + q);
    const float4 h1 = *(const float4*)(hp + (size_t)N * GAT_D + (size_t)node * GAT_D + q);
    float4 o;
    o.x = 0.5f * (h0.x * inv0 + h1.x * inv1);
    o.y = 0.5f * (h0.y * inv0 + h1.y * inv1);
    o.z = 0.5f * (h0.z * inv0 + h1.z * inv1);
    o.w = 0.5f * (h0.w * inv0 + h1.w * inv1);
    *(float4*)(out + (size_t)node * GAT_D + q) = o;
}

// ---------------------------------------------------------------------------
extern "C" void kernel_launch(void* const* d_in, const int* in_sizes, int n_in,
                              void* d_out, int out_size, void* d_ws, size_t ws_size,
                              hipStream_t stream)
{
    const float* x  = (const float*)d_in[0];   // [N, 64]
    const int*   ei = (const int*)  d_in[1];   // [2, E] int32
    const float* W  = (const float*)d_in[2];   // [2, 64]
    const float* a  = (const float*)d_in[3];   // [2, 128]

    const int N = in_sizes[0] / GAT_D;
    const int E = in_sizes[1] / 2;
    const int* row = ei;
    const int* col = ei + E;

    // workspace layout (floats): hp[2*N*64] | rowsum[2*N] | scores[4*N]
    float* ws     = (float*)d_ws;
    float* hp     = ws;
    float* rowsum = ws + (size_t)2 * N * GAT_D;
    float* scores = rowsum + (size_t)2 * N;

    // 1) zero accumulators
    const size_t zeroFloats = (size_t)2 * N * GAT_D + (size_t)2 * N; // multiple of 4
    gat_zero<<<1024, 256, 0, stream>>>((float4*)ws, zeroFloats / 4);

    // 2) WMMA score GEMM: [N,64] x [64,4]
    const int nTiles = (N + 15) / 16;
    const int sBlocks = (nTiles + 7) / 8;
    gat_scores_wmma<<<sBlocks, 256, 0, stream>>>(x, W, a, scores, nTiles);

    // 3) edge scatter (one wave per edge)
    const int eBlocks = (E + 7) / 8;
    gat_edges<<<eBlocks, 256, 0, stream>>>(x, row, col, scores, hp, rowsum, E, N);

    // 4) finalize into d_out
    const int fThreads = N * 16;
    gat_finalize<<<(fThreads + 255) / 256, 256, 0, stream>>>(hp, rowsum, (float*)d_out, N);
}